// NRPOP_89343909691959
// MI455X (gfx1250) — compile-verified
//
#include <hip/hip_runtime.h>
#include <hip/hip_bf16.h>
#include <math.h>

typedef __attribute__((ext_vector_type(2))) float v2f;
typedef __attribute__((ext_vector_type(8))) float v8f;

#define NIT   80000      // items (excluding pad row)
#define DD    128
#define KK    20
#define SAMP  12
#define BB    256
#define SS    50
#define MROWS (BB*SS)    // 12800

__device__ __forceinline__ float wave_sum(float v) {
  v += __shfl_xor(v, 16, 32);
  v += __shfl_xor(v,  8, 32);
  v += __shfl_xor(v,  4, 32);
  v += __shfl_xor(v,  2, 32);
  v += __shfl_xor(v,  1, 32);
  return v;
}
__device__ __forceinline__ float dot4(const float4& a, const float4& b) {
  return a.x*b.x + a.y*b.y + a.z*b.z + a.w*b.w;
}
__device__ __forceinline__ float sigmoidf(float x) { return 1.0f / (1.0f + __expf(-x)); }

// ---------------- tiny init: zero pad row ----------------
__global__ void k_zero(float* __restrict__ p, int n) {
  int i = blockIdx.x * blockDim.x + threadIdx.x;
  if (i < n) p[i] = 0.0f;
}

// ---------------- row-wise normalize: y = x * rsqrt(sum(x^2)+1e-12) ----------------
__global__ void k_rownorm(const float* __restrict__ X, float* __restrict__ Y, int nrows) {
  int lane = threadIdx.x & 31, wid = threadIdx.x >> 5;
  int row = blockIdx.x * 8 + wid;
  if (row >= nrows) return;
  float4 v = ((const float4*)(X + (size_t)row * DD))[lane];
  float s = wave_sum(dot4(v, v));
  float r = rsqrtf(s + 1e-12f);
  v.x *= r; v.y *= r; v.z *= r; v.w *= r;
  ((float4*)(Y + (size_t)row * DD))[lane] = v;
}

// ---------------- capsule routing aggregation (input pre-normalized) ----------------
__global__ void k_routing(const float* __restrict__ XN, const int* __restrict__ adj,
                          float* __restrict__ U, int nrows) {
  int lane = threadIdx.x & 31, wid = threadIdx.x >> 5;
  int row = blockIdx.x * 8 + wid;
  if (row >= nrows) return;
  float4 xs = ((const float4*)(XN + (size_t)row * DD))[lane];
  float4 z[SAMP];
#pragma unroll
  for (int m = 0; m < SAMP; ++m) {
    int nb = adj[(size_t)row * SAMP + m] - 1;
    z[m] = ((const float4*)(XN + (size_t)nb * DD))[lane];
  }
  float4 u = xs;
#pragma unroll
  for (int it = 0; it < 4; ++it) {
    float p[SAMP];
#pragma unroll
    for (int m = 0; m < SAMP; ++m) p[m] = wave_sum(dot4(u, z[m]));
    float mx = p[0];
#pragma unroll
    for (int m = 1; m < SAMP; ++m) mx = fmaxf(mx, p[m]);
    float s = 0.f;
#pragma unroll
    for (int m = 0; m < SAMP; ++m) { p[m] = __expf(p[m] - mx); s += p[m]; }
    float inv = 1.0f / s;
    float4 a = xs;
#pragma unroll
    for (int m = 0; m < SAMP; ++m) {
      float w = p[m] * inv;
      a.x += w * z[m].x; a.y += w * z[m].y; a.z += w * z[m].z; a.w += w * z[m].w;
    }
    float nrm = rsqrtf(wave_sum(dot4(a, a)) + 1e-12f);
    u.x = a.x * nrm; u.y = a.y * nrm; u.z = a.z * nrm; u.w = a.w * nrm;
  }
  ((float4*)(U + (size_t)row * DD))[lane] = u;
}

// ---------------- acc = x + u1 + u2 ; IV = layer_norm(acc) ----------------
__global__ void k_ln_add(const float* __restrict__ X, const float* __restrict__ U1,
                         const float* __restrict__ U2, const float* __restrict__ g,
                         const float* __restrict__ bt, float* __restrict__ IV, int nrows) {
  int lane = threadIdx.x & 31, wid = threadIdx.x >> 5;
  int row = blockIdx.x * 8 + wid;
  if (row >= nrows) return;
  float4 a = ((const float4*)(X + (size_t)row * DD))[lane];
  float4 b = ((const float4*)(U1 + (size_t)row * DD))[lane];
  float4 c = ((const float4*)(U2 + (size_t)row * DD))[lane];
  a.x += b.x + c.x; a.y += b.y + c.y; a.z += b.z + c.z; a.w += b.w + c.w;
  float mu = wave_sum(a.x + a.y + a.z + a.w) * (1.0f / DD);
  float4 d = {a.x - mu, a.y - mu, a.z - mu, a.w - mu};
  float var = wave_sum(dot4(d, d)) * (1.0f / DD);
  float k = rsqrtf(var + 1e-5f);
  float4 gv = ((const float4*)g)[lane];
  float4 bv = ((const float4*)bt)[lane];
  float4 y = {d.x*k*gv.x + bv.x, d.y*k*gv.y + bv.y, d.z*k*gv.z + bv.z, d.w*k*gv.w + bv.w};
  ((float4*)(IV + (size_t)row * DD))[lane] = y;
}

// ---------------- generic row layernorm (for ht / ht_pop) ----------------
__global__ void k_ln_rows(const float* __restrict__ X, const float* __restrict__ g,
                          const float* __restrict__ bt, float* __restrict__ Y, int nrows) {
  int lane = threadIdx.x & 31, wid = threadIdx.x >> 5;
  int row = blockIdx.x * 8 + wid;
  if (row >= nrows) return;
  float4 a = ((const float4*)(X + (size_t)row * DD))[lane];
  float mu = wave_sum(a.x + a.y + a.z + a.w) * (1.0f / DD);
  float4 d = {a.x - mu, a.y - mu, a.z - mu, a.w - mu};
  float var = wave_sum(dot4(d, d)) * (1.0f / DD);
  float k = rsqrtf(var + 1e-5f);
  float4 gv = ((const float4*)g)[lane];
  float4 bv = ((const float4*)bt)[lane];
  float4 y = {d.x*k*gv.x + bv.x, d.y*k*gv.y + bv.y, d.z*k*gv.z + bv.z, d.w*k*gv.w + bv.w};
  ((float4*)(Y + (size_t)row * DD))[lane] = y;
}

// ---------------- paired-k transpose: in[R x 128] -> out2[k/2][R][2] (float2 units) ----------
// out2[(k>>1)*R + n] = { in[n][k], in[n][k+1] }  -> WMMA B fragments become single b64 loads.
__global__ void k_transpose_pk(const float* __restrict__ in, float* __restrict__ out,
                               int R, int C) {
  __shared__ float tile[32][33];
  int x = threadIdx.x;
  int r0 = blockIdx.x * 32, c0 = blockIdx.y * 32;
  for (int yy = threadIdx.y; yy < 32; yy += 8)
    tile[yy][x] = in[(size_t)(r0 + yy) * C + c0 + x];
  __syncthreads();
  float2* out2 = (float2*)out;
  for (int p = threadIdx.y; p < 16; p += 8) {
    float2 v = make_float2(tile[x][2 * p], tile[x][2 * p + 1]);
    out2[(size_t)(c0 / 2 + p) * R + r0 + x] = v;
  }
}

// ---------------- generic WMMA f32 GEMM: C[M x N] = act(Agather @ W^T + bias) ----------------
// K fixed to 128. One wave per 16x16 C tile, 32 x V_WMMA_F32_16X16X4_F32 along K.
// gidx != null: row m -> id = gidx[m]; id==0 -> zero-row pointer (no per-element multiply).
__global__ void __launch_bounds__(256)
k_gemm_wmma(const float* __restrict__ A, const int* __restrict__ gidx,
            const float* __restrict__ zrow,
            const float* __restrict__ W, const float* __restrict__ bias,
            float* __restrict__ C, int ntilesN, int N, int mode) {
  int lane = threadIdx.x & 31, wid = threadIdx.x >> 5;
  int t = blockIdx.x * 8 + wid;
  int mt = t / ntilesN, nt = t % ntilesN;
  int mrow = mt * 16 + (lane & 15);
  int kb = (lane >> 4) * 2;          // A: lanes<16 hold K0/K1, lanes>=16 hold K2/K3
  const float* ar;
  if (gidx) {
    int id = gidx[mrow];
    ar = (id == 0) ? zrow : (A + (size_t)(id - 1) * DD);
  } else {
    ar = A + (size_t)mrow * DD;
  }
  int ncol = nt * 16 + (lane & 15);
  const float* wr = W + (size_t)ncol * DD;   // B_wmma[k][n] = W[n][k]
  v8f c = {0.f,0.f,0.f,0.f,0.f,0.f,0.f,0.f};
#pragma unroll
  for (int kk = 0; kk < 32; ++kk) {
    v2f a = *(const v2f*)(ar + 4 * kk + kb);
    v2f b = *(const v2f*)(wr + 4 * kk + kb);
    c = __builtin_amdgcn_wmma_f32_16x16x4_f32(false, a, false, b, (short)0, c, false, false);
  }
  float bb = bias ? bias[ncol] : 0.0f;
  int rbase = mt * 16 + ((lane >> 4) << 3);
#pragma unroll
  for (int r = 0; r < 8; ++r) {
    float v = c[r] + bb;
    if (mode == 1) v = (v > 0.0f) ? v : 0.01f * v;   // leaky_relu
    C[(size_t)(rbase + r) * N + ncol] = v;
  }
}

// ---------------- fused: logits=h@W2^T, softmax, popvec=p@AE, LN2 -> PV ----------------
__global__ void k_pop(const float* __restrict__ H, const float* __restrict__ IV,
                      const int* __restrict__ anchors, const float* __restrict__ W2,
                      const float* __restrict__ b2, const float* __restrict__ tau,
                      const float* __restrict__ g, const float* __restrict__ bt,
                      float* __restrict__ PV, int nrows) {
  __shared__ float W2s[KK * DD];
  __shared__ float AEs[KK * DD];
  __shared__ float b2s[KK];
  int tid = threadIdx.x, lane = tid & 31, wid = tid >> 5;
  for (int i = tid; i < KK * DD; i += 256) {
    W2s[i] = W2[i];
    int j = i >> 7, k = i & 127;
    AEs[i] = IV[(size_t)(anchors[j] - 1) * DD + k];
  }
  if (tid < KK) b2s[tid] = b2[tid];
  __syncthreads();
  float invtau = 1.0f / tau[0];
  float4 gv = ((const float4*)g)[lane];
  float4 bv = ((const float4*)bt)[lane];
  for (int row = blockIdx.x * 8 + wid; row < nrows; row += gridDim.x * 8) {
    float4 h = ((const float4*)(H + (size_t)row * DD))[lane];
    float p[KK];
#pragma unroll
    for (int j = 0; j < KK; ++j) {
      const float* w = &W2s[j * DD + 4 * lane];
      p[j] = wave_sum(h.x*w[0] + h.y*w[1] + h.z*w[2] + h.w*w[3]);
      p[j] = (p[j] + b2s[j]) * invtau;
    }
    float mx = p[0];
#pragma unroll
    for (int j = 1; j < KK; ++j) mx = fmaxf(mx, p[j]);
    float s = 0.f;
#pragma unroll
    for (int j = 0; j < KK; ++j) { p[j] = __expf(p[j] - mx); s += p[j]; }
    float inv = 1.0f / s;
    float4 a = {0.f, 0.f, 0.f, 0.f};
#pragma unroll
    for (int j = 0; j < KK; ++j) {
      float w = p[j] * inv;
      const float* e = &AEs[j * DD + 4 * lane];
      a.x += w * e[0]; a.y += w * e[1]; a.z += w * e[2]; a.w += w * e[3];
    }
    float mu = wave_sum(a.x + a.y + a.z + a.w) * (1.0f / DD);
    float4 d = {a.x - mu, a.y - mu, a.z - mu, a.w - mu};
    float var = wave_sum(dot4(d, d)) * (1.0f / DD);
    float k = rsqrtf(var + 1e-5f);
    float4 y = {d.x*k*gv.x + bv.x, d.y*k*gv.y + bv.y, d.z*k*gv.z + bv.z, d.w*k*gv.w + bv.w};
    ((float4*)(PV + (size_t)row * DD))[lane] = y;
  }
}

// ---------------- GRU scan: one block per batch row, W_hh^T staged in 192KB LDS ----------------
__global__ void __launch_bounds__(384)
k_gru(const float* __restrict__ GI, const float* __restrict__ Whh,
      const float* __restrict__ bhh, const int* __restrict__ lengths,
      float* __restrict__ HTraw) {
  __shared__ float WT[DD * 384];   // [k][j], 196608 bytes (CDNA5: 320KB LDS/WGP)
  __shared__ float hS[DD];
  __shared__ float ghS[384];
  __shared__ float snap[DD];
  int t = threadIdx.x, b = blockIdx.x;
  for (int i = t; i < DD * 384; i += 384) {
    int j = i >> 7, k = i & 127;
    WT[k * 384 + j] = Whh[i];
  }
  if (t < DD) hS[t] = 0.0f;
  float bh = bhh[t];
  int len = lengths[b];
  const float* gi = GI + (size_t)b * SS * 384;
  __syncthreads();
  for (int s = 0; s < SS; ++s) {
    float acc = bh;
#pragma unroll 8
    for (int k = 0; k < DD; k += 4) {
      float4 hv = *(const float4*)&hS[k];
      acc += hv.x * WT[(k+0)*384 + t] + hv.y * WT[(k+1)*384 + t]
           + hv.z * WT[(k+2)*384 + t] + hv.w * WT[(k+3)*384 + t];
    }
    ghS[t] = acc;
    __syncthreads();
    if (t < DD) {
      float ir = gi[s*384 + t], iz = gi[s*384 + 128 + t], in_ = gi[s*384 + 256 + t];
      float r  = sigmoidf(ir + ghS[t]);
      float zg = sigmoidf(iz + ghS[t + 128]);
      float n  = tanhf(in_ + r * ghS[t + 256]);
      float hn = (1.0f - zg) * n + zg * hS[t];
      hS[t] = hn;
      if (s == len - 1) snap[t] = hn;
    }
    __syncthreads();
  }
  if (t < DD) HTraw[(size_t)b * DD + t] = snap[t];
}

// ---------------- scores: dual WMMA GEMM, 4 M-tiles/wave, paired-k B loads ----------------
__global__ void __launch_bounds__(256)
k_scores(const float* __restrict__ HT, const float* __restrict__ HTP,
         const float* __restrict__ IVT2, const float* __restrict__ PVT2,
         const float* __restrict__ a1, const float* __restrict__ a2,
         float* __restrict__ out) {
  int lane = threadIdx.x & 31, wid = threadIdx.x >> 5;
  int t = blockIdx.x * 8 + wid;        // 20000 waves = 4 Mgroups x 5000 Ntiles
  int g = t & 3, nt = t >> 2;
  int hi = lane >> 4;                  // half-wave selects K pair {0,1} vs {2,3}
  int kb = hi * 2;
  int mr = lane & 15;
  const float* ai[4]; const float* ap[4];
#pragma unroll
  for (int mq = 0; mq < 4; ++mq) {
    int row = (g * 4 + mq) * 16 + mr;
    ai[mq] = HT  + (size_t)row * DD;
    ap[mq] = HTP + (size_t)row * DD;
  }
  int ncol = nt * 16 + mr;
  v8f ci[4], cp[4];
#pragma unroll
  for (int mq = 0; mq < 4; ++mq) {
    ci[mq] = (v8f){0.f,0.f,0.f,0.f,0.f,0.f,0.f,0.f};
    cp[mq] = (v8f){0.f,0.f,0.f,0.f,0.f,0.f,0.f,0.f};
  }
#pragma unroll
  for (int kk = 0; kk < 32; ++kk) {
    size_t pofs = (((size_t)(2 * kk + hi) * NIT + ncol) << 1);
    v2f bi = *(const v2f*)(IVT2 + pofs);     // {IV[n][k0], IV[n][k0+1]} in one b64
    v2f bp = *(const v2f*)(PVT2 + pofs);
#pragma unroll
    for (int mq = 0; mq < 4; ++mq) {
      v2f av = *(const v2f*)(ai[mq] + 4 * kk + kb);
      ci[mq] = __builtin_amdgcn_wmma_f32_16x16x4_f32(false, av, false, bi, (short)0, ci[mq], false, false);
      v2f aw = *(const v2f*)(ap[mq] + 4 * kk + kb);
      cp[mq] = __builtin_amdgcn_wmma_f32_16x16x4_f32(false, aw, false, bp, (short)0, cp[mq], false, false);
    }
  }
  float s1 = sigmoidf(a1[0]), s2 = sigmoidf(a2[0]);
  const size_t NM = (size_t)BB * NIT;
#pragma unroll
  for (int mq = 0; mq < 4; ++mq) {
    int rbase = (g * 4 + mq) * 16 + hi * 8;
#pragma unroll
    for (int r = 0; r < 8; ++r) {
      float vi = ci[mq][r], vp = cp[mq][r];
      size_t o = (size_t)(rbase + r) * NIT + ncol;
      out[o]        = s1 * vi + s2 * vp;   // scores
      out[NM + o]   = vi;                  // scores_item
      out[2*NM + o] = vp;                  // scores_pop
    }
  }
}

extern "C" void kernel_launch(void* const* d_in, const int* in_sizes, int n_in,
                              void* d_out, int out_size, void* d_ws, size_t ws_size,
                              hipStream_t stream) {
  const int*   sess    = (const int*)d_in[0];
  const int*   lengths = (const int*)d_in[1];
  const float* tau     = (const float*)d_in[4];
  const int*   adj     = (const int*)d_in[5];
  const int*   anchors = (const int*)d_in[6];
  const float* emb     = (const float*)d_in[7];
  const float* W_ih    = (const float*)d_in[8];
  const float* W_hh    = (const float*)d_in[9];
  const float* b_ih    = (const float*)d_in[10];
  const float* b_hh    = (const float*)d_in[11];
  const float* W1      = (const float*)d_in[12];
  const float* b1      = (const float*)d_in[13];
  const float* W2      = (const float*)d_in[14];
  const float* b2      = (const float*)d_in[15];
  const float* a1      = (const float*)d_in[16];
  const float* a2      = (const float*)d_in[17];
  const float* ln1g = (const float*)d_in[18]; const float* ln1b = (const float*)d_in[19];
  const float* ln2g = (const float*)d_in[20]; const float* ln2b = (const float*)d_in[21];
  const float* ln3g = (const float*)d_in[22]; const float* ln3b = (const float*)d_in[23];
  const float* ln4g = (const float*)d_in[24]; const float* ln4b = (const float*)d_in[25];

  const size_t TBL = (size_t)NIT * DD;          // 10,240,000 floats
  float* buf0 = (float*)d_ws;                   // XN -> IV
  float* buf1 = buf0 + TBL;                     // U1 -> Htmp -> PVT2
  float* buf2 = buf1 + TBL;                     // U2 -> PV
  float* buf3 = buf2 + TBL;                     // IVT2
  float* gi_i = buf3 + TBL;                     // 12800*384
  float* gi_p = gi_i + (size_t)MROWS * 384;
  float* htraw_i = gi_p + (size_t)MROWS * 384;  // 256*128 each
  float* htraw_p = htraw_i + BB * DD;
  float* ht_i = htraw_p + BB * DD;
  float* ht_p = ht_i + BB * DD;
  float* zrow = ht_p + BB * DD;                 // 128-float zero pad row

  // 0) zero pad row (used for sess index 0 in gathered GEMMs)
  k_zero<<<1, DD, 0, stream>>>(zrow, DD);
  // 1) xn = normalize(emb[1:])
  k_rownorm<<<10000, 256, 0, stream>>>(emb + DD, buf0, NIT);
  // 2-3) two hops of capsule routing
  k_routing<<<10000, 256, 0, stream>>>(buf0, adj + SAMP, buf1, NIT);
  k_routing<<<10000, 256, 0, stream>>>(buf1, adj + SAMP, buf2, NIT);
  // 4) IV = LN1(x + u1 + u2)
  k_ln_add<<<10000, 256, 0, stream>>>(emb + DD, buf1, buf2, ln1g, ln1b, buf0, NIT);
  // 5) IVT2 (paired-k transposed layout)
  k_transpose_pk<<<dim3(NIT/32, DD/32), dim3(32, 8), 0, stream>>>(buf0, buf3, NIT, DD);
  // 6) Htmp = leaky_relu(IV @ W1^T + b1)   [80000x128x128, WMMA]
  k_gemm_wmma<<<5000, 256, 0, stream>>>(buf0, nullptr, zrow, W1, b1, buf1, DD/16, DD, 1);
  // 7) PV = LN2(softmax(Htmp@W2^T/tau) @ anchors_emb)
  k_pop<<<1250, 256, 0, stream>>>(buf1, buf0, anchors, W2, b2, tau, ln2g, ln2b, buf2, NIT);
  // 8) PVT2 (reuse buf1)
  k_transpose_pk<<<dim3(NIT/32, DD/32), dim3(32, 8), 0, stream>>>(buf2, buf1, NIT, DD);
  // 9-10) GI = gather(table)[sess] @ W_ih^T + b_ih   [12800x384x128, WMMA, fused gather]
  k_gemm_wmma<<<2400, 256, 0, stream>>>(buf0, sess, zrow, W_ih, b_ih, gi_i, 384/16, 384, 0);
  k_gemm_wmma<<<2400, 256, 0, stream>>>(buf2, sess, zrow, W_ih, b_ih, gi_p, 384/16, 384, 0);
  // 11-12) GRU scans (one block per batch row, W_hh^T in LDS)
  k_gru<<<BB, 384, 0, stream>>>(gi_i, W_hh, b_hh, lengths, htraw_i);
  k_gru<<<BB, 384, 0, stream>>>(gi_p, W_hh, b_hh, lengths, htraw_p);
  // 13-14) LN3/LN4 on selected hidden states
  k_ln_rows<<<32, 256, 0, stream>>>(htraw_i, ln3g, ln3b, ht_i, BB);
  k_ln_rows<<<32, 256, 0, stream>>>(htraw_p, ln4g, ln4b, ht_p, BB);
  // 15) scores (dual WMMA, 4 M-tiles/wave, 3 outputs)
  k_scores<<<2500, 256, 0, stream>>>(ht_i, ht_p, buf3, buf1, a1, a2, (float*)d_out);
}